// LUTFakeQuant_12257836663001
// MI455X (gfx1250) — compile-verified
//
#include <hip/hip_runtime.h>

// LUT fake-quant, memory-bound streaming formulation for gfx1250 (MI455X).
//
//   t   = clip(x * 128/(scale_c + 1e-8), -128, 127)
//   q   = nearest(round(centers), t)          -> piecewise-constant on 0.5-cells
//   out = q * scale_c / 128
//
// Since round(centers) are integers, decision boundaries are half-integers,
// so the whole quantizer over the clipped domain is a 512-entry LUT indexed
// by j = (int)clip(2*t + 256, 0, 510).  Fold 2/(scale+eps)*128 = 256/(scale+eps)
// into one per-channel constant so the index is a single fma+med3+cvt.

#define THREADS 256
#define BLOCKS  2048

// d_ws float layout: [0..511] lut, [512..639] inv2 = 256/(s+eps), [640..767] osc = s/128
__global__ __launch_bounds__(THREADS) void lutfq_setup_kernel(
    const float* __restrict__ scale,
    const float* __restrict__ centers,
    float* __restrict__ ws)
{
    __shared__ float s_c[16];
    const int t = threadIdx.x;
    if (t < 16) s_c[t] = rintf(centers[t]);          // jnp.round = RNE
    __syncthreads();

    // Each LUT cell e covers t in [(e-256)/2, (e-255)/2); nearest center is
    // constant on the cell interior -> evaluate at the cell midpoint.
    for (int e = t; e < 512; e += THREADS) {
        const float p = (float)(e - 256) * 0.5f + 0.25f;
        float best = s_c[0];
        float bd   = fabsf(p - best);
#pragma unroll
        for (int k = 1; k < 16; ++k) {
            const float c = s_c[k];
            const float d = fabsf(p - c);
            if (d < bd) { bd = d; best = c; }        // first-wins on ties
        }
        ws[e] = best;
    }
    if (t < 128) {
        const float s = scale[t];
        ws[512 + t] = 256.0f / (s + 1e-8f);          // index scale (already x2)
        ws[640 + t] = s * (1.0f / 128.0f);           // output rescale
    }
}

__global__ __launch_bounds__(THREADS) void lutfq_main_kernel(
    const float* __restrict__ x,
    const float* __restrict__ ws,
    float* __restrict__ out,
    int n4, int ntail)
{
    __shared__ float s_lut[512];
    const int t = threadIdx.x;
    // 2 KB LUT: global -> LDS, 2 dwords/thread
    s_lut[t]       = ws[t];
    s_lut[t + 256] = ws[t + 256];
    __syncthreads();

    const float* __restrict__ inv2 = ws + 512;
    const float* __restrict__ osc  = ws + 640;

    const int i0     = blockIdx.x * THREADS + t;
    const int stride = THREADS * BLOCKS;             // multiple of 32 groups

    // 128 channels / 4-per-float4 = 32 groups per channel row; stride%32==0
    // -> each thread's channel quad is loop-invariant: hoist its constants.
    const int cg = (i0 & 31) << 2;
    const float4 iv = *(const float4*)(inv2 + cg);
    const float4 ov = *(const float4*)(osc  + cg);

    const float4* __restrict__ x4 = (const float4*)x;
    float4* __restrict__ o4       = (float4*)out;

    for (int i = i0; i < n4; i += stride) {
        if (i + stride < n4)
            __builtin_prefetch(x4 + i + stride, 0, 1);   // global_prefetch_b8
        const float4 xv = x4[i];                          // global_load_b128
        float4 r;
        {
            const float u = fminf(fmaxf(fmaf(xv.x, iv.x, 256.0f), 0.0f), 510.0f);
            r.x = s_lut[(int)u] * ov.x;                   // ds_load_b32
        }
        {
            const float u = fminf(fmaxf(fmaf(xv.y, iv.y, 256.0f), 0.0f), 510.0f);
            r.y = s_lut[(int)u] * ov.y;
        }
        {
            const float u = fminf(fmaxf(fmaf(xv.z, iv.z, 256.0f), 0.0f), 510.0f);
            r.z = s_lut[(int)u] * ov.z;
        }
        {
            const float u = fminf(fmaxf(fmaf(xv.w, iv.w, 256.0f), 0.0f), 510.0f);
            r.w = s_lut[(int)u] * ov.w;
        }
        o4[i] = r;                                        // global_store_b128
    }

    // Scalar tail (n % 4 != 0); zero iterations for the reference shape.
    if (i0 < ntail) {
        const int idx = (n4 << 2) + i0;
        const int c   = idx & 127;
        const float u = fminf(fmaxf(fmaf(x[idx], inv2[c], 256.0f), 0.0f), 510.0f);
        out[idx] = s_lut[(int)u] * osc[c];
    }
}

extern "C" void kernel_launch(void* const* d_in, const int* in_sizes, int n_in,
                              void* d_out, int out_size, void* d_ws, size_t ws_size,
                              hipStream_t stream) {
    const float* x       = (const float*)d_in[0];   // (16,64,64,128) f32
    const float* scale   = (const float*)d_in[1];   // (128,) f32
    const float* centers = (const float*)d_in[2];   // (16,)  f32
    float* out = (float*)d_out;
    float* ws  = (float*)d_ws;                      // needs 768 floats = 3 KB

    const int n     = in_sizes[0];
    const int n4    = n >> 2;
    const int ntail = n - (n4 << 2);

    lutfq_setup_kernel<<<1, THREADS, 0, stream>>>(scale, centers, ws);

    int blocks = (n4 + THREADS - 1) / THREADS;
    if (blocks > BLOCKS) blocks = BLOCKS;
    if (blocks < 1) blocks = 1;
    lutfq_main_kernel<<<blocks, THREADS, 0, stream>>>(x, ws, out, n4, ntail);
}